// MessagePassing_50465865728371
// MI455X (gfx1250) — compile-verified
//
#include <hip/hip_runtime.h>

#define NN 50000
#define NE 800000
#define DD 64

typedef float v2f __attribute__((ext_vector_type(2)));
typedef float v8f __attribute__((ext_vector_type(8)));

// ---------------- range / denom prep (deterministic, no atomics) -------------

__global__ void fill_zero_i32(int* __restrict__ p, int n) {
  int i = blockIdx.x * blockDim.x + threadIdx.x;
  if (i < n) p[i] = 0;
}

// edge_dst is sorted: mark segment boundaries
__global__ void find_boundaries(const int* __restrict__ dst,
                                int* __restrict__ rb, int* __restrict__ re, int E) {
  int e = blockIdx.x * blockDim.x + threadIdx.x;
  if (e >= E) return;
  int d = dst[e];
  if (e == 0 || dst[e - 1] != d) rb[d] = e;
  if (e == E - 1 || dst[e + 1] != d) re[d] = e + 1;
}

__global__ void compute_inv_denom(const float* __restrict__ w,
                                  const int* __restrict__ rb, const int* __restrict__ re,
                                  float* __restrict__ inv_denom, int n) {
  int i = blockIdx.x * blockDim.x + threadIdx.x;
  if (i >= n) return;
  float s = 0.0f;
  int b = rb[i], e = re[i];
  for (int k = b; k < e; ++k) s += w[k];
  inv_denom[i] = 1.0f / (s + 1e-9f);
}

// ---------------- gather + segmented sum: one wave32 per node ----------------

__global__ void aggregate(const float* __restrict__ feats,
                          const int* __restrict__ src, const float* __restrict__ w,
                          const int* __restrict__ rb, const int* __restrict__ re,
                          const float* __restrict__ inv_denom,
                          float* __restrict__ msg) {
  int wave = blockIdx.x * (blockDim.x >> 5) + (threadIdx.x >> 5);
  int lane = threadIdx.x & 31;
  if (wave >= NN) return;                 // wave-uniform exit
  int b = rb[wave], e = re[wave];
  float inv = inv_denom[wave];
  int col = lane * 2;
  float ax = 0.0f, ay = 0.0f;
  for (int i = b; i < e; ++i) {
    int s = src[i];
    float ww = w[i];
    if (i + 1 < e) {
      int sn = src[i + 1];
      __builtin_prefetch(feats + (size_t)sn * DD + col, 0, 0);  // global_prefetch_b8
    }
    v2f f = *(const v2f*)(feats + (size_t)s * DD + col);
    ax = fmaf(ww, f.x, ax);
    ay = fmaf(ww, f.y, ay);
  }
  v2f out;
  out.x = ax * inv;
  out.y = ay * inv;
  *(v2f*)(msg + (size_t)wave * DD + col) = out;
}

// ----- fused dual-GEMM + bias + ReLU via V_WMMA_F32_16X16X4_F32 --------------
// out = relu(feats @ W0^T + b0 + msg @ W1^T + b1), one wave per 16-row tile.

__global__ void gemm_relu(const float* __restrict__ feats, const float* __restrict__ msg,
                          const float* __restrict__ W0, const float* __restrict__ b0,
                          const float* __restrict__ W1, const float* __restrict__ b1,
                          float* __restrict__ out) {
  const int ntiles = NN / 16;             // 3125
  int wave = blockIdx.x * (blockDim.x >> 5) + (threadIdx.x >> 5);
  int lane = threadIdx.x & 31;
  if (wave >= ntiles) return;             // wave-uniform: EXEC stays all-1 for WMMA
  int row0 = wave * 16;
  int mrow = row0 + (lane & 15);
  int khalf = (lane >> 4) * 2;            // lanes 0-15: K=k0,k0+1 ; lanes 16-31: K=k0+2,k0+3

  // A fragments for all 16 K-steps (ISA 16x4 f32 A layout), both matrices.
  v2f af[16], am[16];
#pragma unroll
  for (int k = 0; k < 16; ++k) {
    af[k] = *(const v2f*)(feats + (size_t)mrow * DD + 4 * k + khalf);
    am[k] = *(const v2f*)(msg   + (size_t)mrow * DD + 4 * k + khalf);
  }

  int nsub = lane & 15;
#pragma unroll
  for (int jt = 0; jt < 4; ++jt) {
    int j = jt * 16 + nsub;               // output column this lane owns in the tile
    float bias = b0[j] + b1[j];
    v8f acc = {bias, bias, bias, bias, bias, bias, bias, bias};
#pragma unroll
    for (int k = 0; k < 16; ++k) {
      // B[k][j] = W[j][k]  (x @ W^T), matching 4x16 f32 B fragment layout
      v2f bw0 = *(const v2f*)(W0 + (size_t)j * DD + 4 * k + khalf);
      v2f bw1 = *(const v2f*)(W1 + (size_t)j * DD + 4 * k + khalf);
      acc = __builtin_amdgcn_wmma_f32_16x16x4_f32(false, af[k], false, bw0,
                                                  (short)0, acc, false, false);
      acc = __builtin_amdgcn_wmma_f32_16x16x4_f32(false, am[k], false, bw1,
                                                  (short)0, acc, false, false);
    }
    // C/D layout: VGPR i, lanes<16 -> M=i, lanes>=16 -> M=i+8; N = lane&15
#pragma unroll
    for (int i = 0; i < 8; ++i) {
      float v = acc[i] > 0.0f ? acc[i] : 0.0f;
      int r = row0 + i + ((lane >> 4) << 3);
      out[(size_t)r * DD + jt * 16 + nsub] = v;
    }
  }
}

// ---------------------------------------------------------------------------

extern "C" void kernel_launch(void* const* d_in, const int* in_sizes, int n_in,
                              void* d_out, int out_size, void* d_ws, size_t ws_size,
                              hipStream_t stream) {
  const float* node_feats = (const float*)d_in[0];
  const int*   edge_src   = (const int*)d_in[1];
  const int*   edge_dst   = (const int*)d_in[2];
  const float* edge_w     = (const float*)d_in[3];
  const float* W0         = (const float*)d_in[4];
  const float* b0         = (const float*)d_in[5];
  const float* W1         = (const float*)d_in[6];
  const float* b1         = (const float*)d_in[7];
  float* out = (float*)d_out;

  // workspace layout
  float* msg       = (float*)d_ws;                       // NN*DD f32
  float* ftmp      = msg + (size_t)NN * DD;              // NN*DD f32
  int*   rb        = (int*)(ftmp + (size_t)NN * DD);     // NN i32
  int*   re        = rb + NN;                            // NN i32 (contiguous with rb)
  float* inv_denom = (float*)(re + NN);                  // NN f32

  const int B = 256;
  fill_zero_i32<<<(2 * NN + B - 1) / B, B, 0, stream>>>(rb, 2 * NN);
  find_boundaries<<<(NE + B - 1) / B, B, 0, stream>>>(edge_dst, rb, re, NE);
  compute_inv_denom<<<(NN + B - 1) / B, B, 0, stream>>>(edge_w, rb, re, inv_denom, NN);

  const int wpb = B / 32;                                // 8 waves per block
  int agg_blocks  = (NN + wpb - 1) / wpb;
  int ntiles      = NN / 16;
  int gemm_blocks = (ntiles + wpb - 1) / wpb;

  // layer 0: node_feats -> ftmp
  aggregate<<<agg_blocks, B, 0, stream>>>(node_feats, edge_src, edge_w, rb, re, inv_denom, msg);
  gemm_relu<<<gemm_blocks, B, 0, stream>>>(node_feats, msg, W0, b0, W1, b1, ftmp);

  // layer 1: ftmp -> out
  aggregate<<<agg_blocks, B, 0, stream>>>(ftmp, edge_src, edge_w, rb, re, inv_denom, msg);
  gemm_relu<<<gemm_blocks, B, 0, stream>>>(ftmp, msg, W0, b0, W1, b1, out);
}